// Encoder_59279138619817
// MI455X (gfx1250) — compile-verified
//
#include <hip/hip_runtime.h>
#include <hip/hip_bf16.h>

#define DIM 128

typedef __attribute__((ext_vector_type(2))) float v2f;
typedef __attribute__((ext_vector_type(8))) float v8f;

// ---------------------------------------------------------------------------
// Degree count per destination node (float so we can reuse buffer for 1/cnt).
// ---------------------------------------------------------------------------
__global__ __launch_bounds__(256) void count_kernel(const long long* __restrict__ dst,
                                                    float* __restrict__ cnt, int E) {
  int e = blockIdx.x * blockDim.x + threadIdx.x;
  if (e < E) atomicAdd(&cnt[(int)dst[e]], 1.0f);
}

__global__ __launch_bounds__(256) void inv_kernel(float* __restrict__ cnt, int n) {
  int i = blockIdx.x * blockDim.x + threadIdx.x;
  if (i < n) cnt[i] = 1.0f / fmaxf(cnt[i], 1.0f);
}

// ---------------------------------------------------------------------------
// Edge scatter: agg[dst] += relu(z[src] + edge_attr).  32 threads per edge,
// each handles one float4 chunk -> coalesced 512B per edge per wave, then
// 4x global_atomic_add_f32 into the L2-resident accumulator (25.6 MB << 192MB).
// ---------------------------------------------------------------------------
__global__ __launch_bounds__(256) void scatter_kernel(const float* __restrict__ z,
                                                      const long long* __restrict__ src,
                                                      const long long* __restrict__ dst,
                                                      const float* __restrict__ ea,
                                                      float* __restrict__ agg, int E) {
  long long idx = (long long)blockIdx.x * blockDim.x + threadIdx.x;
  int e = (int)(idx >> 5);
  if (e >= E) return;
  int c = ((int)idx & 31) * 4;
  long long s = src[e];
  long long d = dst[e];
  const float4 xv = *(const float4*)(z + s * DIM + c);
  const float4 ev = *(const float4*)(ea + (long long)e * DIM + c);
  float4 m;
  m.x = fmaxf(xv.x + ev.x, 0.0f);
  m.y = fmaxf(xv.y + ev.y, 0.0f);
  m.z = fmaxf(xv.z + ev.z, 0.0f);
  m.w = fmaxf(xv.w + ev.w, 0.0f);
  float* ap = agg + d * DIM + c;
  atomicAdd(ap + 0, m.x);
  atomicAdd(ap + 1, m.y);
  atomicAdd(ap + 2, m.z);
  atomicAdd(ap + 3, m.w);
}

// ---------------------------------------------------------------------------
// Fused SAGE linear:  out = (agg * inv) @ Wl + z @ Wr + bl  [optional relu]
// Block = 256 threads = 8 wave32s; block computes a 16-row x 128-col tile.
// Wave w owns the 16-column tile n0 = 16*w.  K accumulated with
// V_WMMA_F32_16X16X4_F32 (exact fp32, matches reference numerics).
//
// LDS row slabs use stride 132 floats: A-frag float2 reads hit banks
// (4m + kk) .. (4m + kk + 3) per lane -> conflict-free across both halves.
// ---------------------------------------------------------------------------
#define SROW 132  // 128 + 4 padding floats

__global__ __launch_bounds__(256) void sage_gemm_kernel(
    const float* __restrict__ agg, const float* __restrict__ inv,
    const float* __restrict__ z,
    const float* __restrict__ Wl, const float* __restrict__ bl,
    const float* __restrict__ Wr,
    float* __restrict__ out, int nrows, int relu_out) {
  __shared__ float sA[16 * SROW];  // scaled agg rows
  __shared__ float sZ[16 * SROW];  // root (z) rows

  const int tid  = threadIdx.x;
  const int wave = tid >> 5;
  const int lane = tid & 31;
  const int r0   = blockIdx.x * 16;

  // Cooperative stage of the 16 input rows (agg pre-scaled by 1/deg).
  for (int i = tid; i < 16 * DIM; i += 256) {
    int r = i >> 7;          // row within slab
    int k = i & (DIM - 1);   // feature index
    int row = r0 + r;
    if (row >= nrows) row = nrows - 1;  // harmless duplicate for tail
    sA[r * SROW + k] = agg[(long long)row * DIM + k] * inv[row];
    sZ[r * SROW + k] = z[(long long)row * DIM + k];
  }
  __syncthreads();

  // ISA 16x4 F32 A-layout: lane l -> row l&15, K pair (l>>4)*2 + {0,1}.
  // B-layout mirrors with N = l&15.
  const int n0   = wave * 16;
  const int m    = lane & 15;
  const int koff = (lane >> 4) * 2;

  const float* WlB = Wl + n0 + m;  // column (n0+m) of Wl, row-major KxN
  const float* WrB = Wr + n0 + m;

  v8f acc = {};

  #pragma unroll 4
  for (int k = 0; k < DIM; k += 4) {
    int kk = k + koff;
    v2f a = *(const v2f*)(sA + m * SROW + kk);
    v2f b;
    b.x = WlB[(size_t)kk * DIM];
    b.y = WlB[(size_t)(kk + 1) * DIM];
    acc = __builtin_amdgcn_wmma_f32_16x16x4_f32(
        /*neg_a=*/false, a, /*neg_b=*/false, b,
        /*c_mod=*/(short)0, acc, /*reuse_a=*/false, /*reuse_b=*/false);
  }

  #pragma unroll 4
  for (int k = 0; k < DIM; k += 4) {
    int kk = k + koff;
    v2f a = *(const v2f*)(sZ + m * SROW + kk);
    v2f b;
    b.x = WrB[(size_t)kk * DIM];
    b.y = WrB[(size_t)(kk + 1) * DIM];
    acc = __builtin_amdgcn_wmma_f32_16x16x4_f32(
        false, a, false, b, (short)0, acc, false, false);
  }

  // C/D layout: VGPR g -> M = g (lanes 0-15) / M = g+8 (lanes 16-31), N = lane&15.
  const float bias = bl[n0 + m];
  const int   ncol = n0 + m;
  const int  mbase = r0 + (lane >> 4) * 8;
  #pragma unroll
  for (int g = 0; g < 8; ++g) {
    int row = mbase + g;
    if (row < nrows) {
      float v = acc[g] + bias;
      if (relu_out) v = fmaxf(v, 0.0f);
      out[(long long)row * DIM + ncol] = v;
    }
  }
}

// ---------------------------------------------------------------------------
// Orchestration (graph-capture safe: only async ops on `stream`).
// ---------------------------------------------------------------------------
extern "C" void kernel_launch(void* const* d_in, const int* in_sizes, int n_in,
                              void* d_out, int out_size, void* d_ws, size_t ws_size,
                              hipStream_t stream) {
  const float*     x   = (const float*)d_in[0];
  const long long* ei  = (const long long*)d_in[1];  // int64 edge_index [2, E]
  const float*     ea  = (const float*)d_in[2];
  const float*     W1l = (const float*)d_in[3];
  const float*     b1l = (const float*)d_in[4];
  const float*     W1r = (const float*)d_in[5];
  const float*     W2l = (const float*)d_in[6];
  const float*     b2l = (const float*)d_in[7];
  const float*     W2r = (const float*)d_in[8];
  float* out = (float*)d_out;

  const int N = in_sizes[0] / DIM;
  const int E = in_sizes[1] / 2;
  const long long* src = ei;       // row 0
  const long long* dst = ei + E;   // row 1

  // Workspace layout: agg[N*DIM] | inv[N] | h1[N*DIM]   (~51.4 MB)
  float* agg = (float*)d_ws;
  float* inv = agg + (size_t)N * DIM;
  float* h1  = inv + N;

  // Degree -> reciprocal (dst is the same for both layers; compute once).
  hipMemsetAsync(inv, 0, (size_t)N * sizeof(float), stream);
  count_kernel<<<(E + 255) / 256, 256, 0, stream>>>(dst, inv, E);
  inv_kernel<<<(N + 255) / 256, 256, 0, stream>>>(inv, N);

  const int rowTiles = (N + 15) / 16;
  const long long sthreads = (long long)E * 32;
  const int sblocks = (int)((sthreads + 255) / 256);

  // ---- Layer 1 ----
  hipMemsetAsync(agg, 0, (size_t)N * DIM * sizeof(float), stream);
  scatter_kernel<<<sblocks, 256, 0, stream>>>(x, src, dst, ea, agg, E);
  sage_gemm_kernel<<<rowTiles, 256, 0, stream>>>(agg, inv, x, W1l, b1l, W1r,
                                                 h1, N, /*relu=*/1);

  // ---- Layer 2 ----
  hipMemsetAsync(agg, 0, (size_t)N * DIM * sizeof(float), stream);
  scatter_kernel<<<sblocks, 256, 0, stream>>>(h1, src, dst, ea, agg, E);
  sage_gemm_kernel<<<rowTiles, 256, 0, stream>>>(agg, inv, h1, W2l, b2l, W2r,
                                                 out, N, /*relu=*/0);
}